// ROIAlign_14869176779326
// MI455X (gfx1250) — compile-verified
//
#include <hip/hip_runtime.h>
#include <cmath>

// ROI Align (torchvision aligned=False), fp32, NCHW. B=2,C=256,H=200,W=304,P=7,s=2,scale=0.25.
//
// MI455X strategy:
//  * features (124.5 MB) < 192 MB L2 -> gathers are L2-resident after first touch.
//  * output (50 MB) is write-once streaming -> nontemporal stores (gfx1250 TH=NT)
//    so it never evicts the feature working set.
//  * horizontal neighbor pairs (xl, xl+1) fetched as one b64 load (edge handled by
//    zeroed weight + lane select) -> 8 global_load_b64 per output instead of 16 b32.
//  * per-ROI coefficients are channel-invariant -> packed float4 per axis-sample in
//    LDS (one ds_load_b128 each), row offsets pre-multiplied (no IMUL in inner loop).

constexpr int   FC     = 256;
constexpr int   FH     = 200;
constexpr int   FW     = 304;
constexpr int   POOLP  = 7;
constexpr int   SRATIO = 2;
constexpr int   PS     = POOLP * SRATIO;   // 14 samples per axis
constexpr int   BINS   = POOLP * POOLP;    // 49
constexpr int   CHUNK  = 64;               // channels per block (grid.y = FC/CHUNK)
constexpr float SCALE  = 0.25f;

typedef float f2v  __attribute__((ext_vector_type(2)));
typedef f2v   f2vu __attribute__((aligned(4)));   // 8B load at 4B alignment

__global__ __launch_bounds__(256) void roi_align_nt_kernel(
    const float* __restrict__ feat,   // [B, FC, FH, FW]
    const float* __restrict__ rois,   // [N, 5] = (b, x1, y1, x2, y2)
    float* __restrict__ out)          // [N, FC, 7, 7]
{
    // y-sample i: { int(lo*FW), int(hi*FW), wy0, wy1 }   (weights zeroed if invalid)
    // x-sample i: { int(base),  int(sel),   wx0, wx1 }   base=min(lo,FW-2), sel=lo-base
    __shared__ float4 s_y[PS];
    __shared__ float4 s_x[PS];
    __shared__ int    s_b;

    const int n   = blockIdx.x;
    const int tid = threadIdx.x;

    if (tid == 0) s_b = (int)rois[(size_t)n * 5];

    if (tid < 2 * PS) {
        const int axis = tid / PS;          // 0 = y, 1 = x
        const int i    = tid % PS;          // sample index along axis
        const float c1 = rois[(size_t)n * 5 + (axis ? 1 : 2)] * SCALE;
        const float c2 = rois[(size_t)n * 5 + (axis ? 3 : 4)] * SCALE;
        const int  size = axis ? FW : FH;

        const float roi_len = fmaxf(c2 - c1, 1.0f);
        const float bin     = roi_len / (float)POOLP;
        const float off     = ((float)i + 0.5f) / (float)SRATIO;
        float c = c1 + off * bin;

        const bool valid = (c >= -1.0f) && (c <= (float)size);
        c = fmaxf(c, 0.0f);
        int lo = (int)floorf(c);
        const bool at_edge = lo >= size - 1;
        lo = min(lo, size - 1);
        const int hi = min(lo + 1, size - 1);
        if (at_edge) c = (float)lo;
        const float frac = c - (float)lo;

        const float w1 = valid ? frac : 0.0f;
        const float w0 = valid ? (1.0f - frac) : 0.0f;

        float4 pk;
        if (axis == 0) {                    // y: pre-multiplied row offsets
            pk.x = __int_as_float(lo * FW);
            pk.y = __int_as_float(hi * FW);
        } else {                            // x: paired-load base + lane select
            const int base = min(lo, FW - 2);
            pk.x = __int_as_float(base);
            pk.y = __int_as_float(lo - base);   // 0 or 1 (1 only when w1==0)
        }
        pk.z = w0;
        pk.w = w1;
        if (axis == 0) s_y[i] = pk; else s_x[i] = pk;
    }
    __syncthreads();

    const size_t plane = (size_t)FH * FW;
    const int    cbase = blockIdx.y * CHUNK;
    const float* fb    = feat + (size_t)s_b * FC * plane + (size_t)cbase * plane;
    float*       ob    = out + ((size_t)n * FC + cbase) * BINS;

    // t = c_local * 49 + bin -> consecutive threads: coalesced NT stores, gathers
    // stay inside one channel plane (L2 locality).
    for (int t = tid; t < CHUNK * BINS; t += 256) {
        const int bin = t % BINS;
        const int cl  = t / BINS;
        const int ph  = bin / POOLP;
        const int pw  = bin % POOLP;

        const float* fp = fb + (size_t)cl * plane;
        float acc = 0.0f;

#pragma unroll
        for (int iy = 0; iy < SRATIO; ++iy) {
            const float4 ys  = s_y[ph * SRATIO + iy];
            const int    or0 = __float_as_int(ys.x);
            const int    or1 = __float_as_int(ys.y);
            const float  wy0 = ys.z, wy1 = ys.w;
#pragma unroll
            for (int ix = 0; ix < SRATIO; ++ix) {
                const float4 xs  = s_x[pw * SRATIO + ix];
                const int    xb  = __float_as_int(xs.x);
                const int    sel = __float_as_int(xs.y);
                const float  wx0 = xs.z, wx1 = xs.w;

                const f2v a = *(const f2vu*)(fp + or0 + xb);   // row lo: v[base], v[base+1]
                const f2v b = *(const f2vu*)(fp + or1 + xb);   // row hi
                const float va_lo = sel ? a.y : a.x;           // v[xl]
                const float vb_lo = sel ? b.y : b.x;
                // v_hi = element .y: equals v[xl+1]=v[xh] when sel==0; when sel==1
                // (right edge) wx1==0 so the value is irrelevant.
                acc += wy0 * (wx0 * va_lo + wx1 * a.y)
                     + wy1 * (wx0 * vb_lo + wx1 * b.y);
            }
        }

        // Streaming output: NT store keeps the 192MB L2 for the feature tensor.
        __builtin_nontemporal_store(acc * 0.25f, &ob[(size_t)cl * BINS + bin]);
    }
}

extern "C" void kernel_launch(void* const* d_in, const int* in_sizes, int n_in,
                              void* d_out, int out_size, void* d_ws, size_t ws_size,
                              hipStream_t stream) {
    const float* feat = (const float*)d_in[0];   // [2,256,200,304] fp32
    const float* rois = (const float*)d_in[1];   // [N,5] fp32
    float*       out  = (float*)d_out;           // [N,256,7,7] fp32

    const int n_rois = in_sizes[1] / 5;          // 1000
    dim3 grid((unsigned)n_rois, FC / CHUNK);     // (1000, 4)
    roi_align_nt_kernel<<<grid, 256, 0, stream>>>(feat, rois, out);
}